// NEGLoss_30614526886301
// MI455X (gfx1250) — compile-verified
//
#include <hip/hip_runtime.h>

// NEG loss, algebraically reduced:
//   negatives are sampled from the distribution with positives masked out,
//   so they never collide with target indices => weights[target] == pos_counts[target].
//   loss = - sum_i c(t_i) * input[i, t_i] / sum_i c(t_i),  c(v) = count of v in target.
// freqs and the PRNG are provably dead. Only 4096 gathered floats of `input` are read.

#define VOCAB 50257
#define BATCH 4096
#define TPB   256
#define NBLK  (BATCH / TPB)   // 16

typedef __attribute__((ext_vector_type(2))) float v2f;
typedef __attribute__((ext_vector_type(8))) float v8f;

// Wave32 sum via the matrix unit: A(16x4) holds the 32 lane values (one per lane in
// VGPR0 slot, 0 in the other), B = ones(4x16). Every column of D = A*B contains the
// 16 row-sums; summing the 8 C/D VGPRs per lane and adding the lane^16 partner gives
// the full 32-lane sum in every lane. Fixed RNE rounding -> deterministic.
__device__ __forceinline__ float wave_sum(float v) {
#if __has_builtin(__builtin_amdgcn_wmma_f32_16x16x4_f32)
  v2f a; a[0] = v;    a[1] = 0.0f;
  v2f b; b[0] = 1.0f; b[1] = 1.0f;
  v8f c = {};
  c = __builtin_amdgcn_wmma_f32_16x16x4_f32(false, a, false, b, (short)0, c, false, false);
  float s = ((c[0] + c[1]) + (c[2] + c[3])) + ((c[4] + c[5]) + (c[6] + c[7]));
  s += __shfl_xor(s, 16, 32);
  return s;
#else
  for (int off = 16; off > 0; off >>= 1) v += __shfl_xor(v, off, 32);
  return v;
#endif
}

// Deterministic block reduction: WMMA wave sums, then wave 0 combines the 8 wave
// partials in a fixed order.
__device__ __forceinline__ void block_reduce_write(float n, float d, float* partials) {
  __shared__ float wn[TPB / 32], wd[TPB / 32];
  float sn = wave_sum(n);
  float sd = wave_sum(d);
  int lane = threadIdx.x & 31;
  int wid  = threadIdx.x >> 5;
  if (lane == 0) { wn[wid] = sn; wd[wid] = sd; }
  __syncthreads();
  if (threadIdx.x == 0) {
    float tn = 0.0f, td = 0.0f;
    for (int w = 0; w < TPB / 32; ++w) { tn += wn[w]; td += wd[w]; }
    partials[2 * blockIdx.x]     = tn;
    partials[2 * blockIdx.x + 1] = td;
  }
}

__global__ void k_init(int* __restrict__ counts) {
  int i = blockIdx.x * blockDim.x + threadIdx.x;
  if (i < VOCAB) counts[i] = 0;
}

__global__ void k_count(const int* __restrict__ target, int* __restrict__ counts) {
  int i = blockIdx.x * blockDim.x + threadIdx.x;
  if (i < BATCH) atomicAdd(&counts[target[i]], 1);
}

__global__ void k_reduce(const float* __restrict__ input, const int* __restrict__ target,
                         const int* __restrict__ counts, float* __restrict__ partials) {
  __shared__ int s_t[TPB];
  const int tid = threadIdx.x;
  const int gi  = blockIdx.x * TPB + tid;

  // Stage this block's target tile into LDS via the CDNA5 async global->LDS path.
#if __has_builtin(__builtin_amdgcn_global_load_async_to_lds_b32)
  __builtin_amdgcn_global_load_async_to_lds_b32(
      (__attribute__((address_space(1))) int*)(target + gi),
      (__attribute__((address_space(3))) int*)(&s_t[tid]),
      0, 0);
#if __has_builtin(__builtin_amdgcn_s_wait_asynccnt)
  __builtin_amdgcn_s_wait_asynccnt(0);
#else
  asm volatile("s_wait_asynccnt 0" ::: "memory");
#endif
#else
  s_t[tid] = target[gi];
#endif
  __syncthreads();

  const int t = s_t[tid];
  const float c = (float)counts[t];
  const float x = input[(size_t)gi * VOCAB + t];
  block_reduce_write(c * x, c, partials);
}

// Fallback when the workspace cannot hold the 50257-entry histogram:
// compute each thread's multiplicity by scanning target through LDS tiles (O(B^2/block)).
__global__ void k_reduce_nocnt(const float* __restrict__ input, const int* __restrict__ target,
                               float* __restrict__ partials) {
  __shared__ int tile[TPB];
  const int tid = threadIdx.x;
  const int gi  = blockIdx.x * TPB + tid;
  const int myt = target[gi];
  int cnt = 0;
  for (int base = 0; base < BATCH; base += TPB) {
    __syncthreads();
    tile[tid] = target[base + tid];
    __syncthreads();
#pragma unroll 8
    for (int j = 0; j < TPB; ++j) cnt += (tile[j] == myt) ? 1 : 0;
  }
  const float c = (float)cnt;
  const float x = input[(size_t)gi * VOCAB + myt];
  block_reduce_write(c * x, c, partials);
}

__global__ void k_final(const float* __restrict__ partials, float* __restrict__ out) {
  float n = 0.0f, d = 0.0f;
  for (int b = 0; b < NBLK; ++b) { n += partials[2 * b]; d += partials[2 * b + 1]; }
  out[0] = -(n / d);
}

extern "C" void kernel_launch(void* const* d_in, const int* in_sizes, int n_in,
                              void* d_out, int out_size, void* d_ws, size_t ws_size,
                              hipStream_t stream) {
  (void)in_sizes; (void)n_in; (void)out_size;
  const float* input  = (const float*)d_in[0];
  const int*   target = (const int*)d_in[2];   // d_in[1] (freqs) provably unused
  float*       out    = (float*)d_out;

  const size_t counts_bytes = (size_t)VOCAB * sizeof(int);
  const size_t part_off     = (counts_bytes + 255) & ~(size_t)255;
  const size_t need         = part_off + (size_t)(2 * NBLK) * sizeof(float);

  if (ws_size >= need) {
    int*   counts   = (int*)d_ws;
    float* partials = (float*)((char*)d_ws + part_off);
    k_init  <<<(VOCAB + TPB - 1) / TPB, TPB, 0, stream>>>(counts);
    k_count <<<NBLK, TPB, 0, stream>>>(target, counts);
    k_reduce<<<NBLK, TPB, 0, stream>>>(input, target, counts, partials);
    k_final <<<1, 1, 0, stream>>>(partials, out);
  } else {
    float* partials = (float*)d_ws;  // 128 bytes
    k_reduce_nocnt<<<NBLK, TPB, 0, stream>>>(input, target, partials);
    k_final       <<<1, 1, 0, stream>>>(partials, out);
  }
}